// MaskCostVolumeLoss_377957122734
// MI455X (gfx1250) — compile-verified
//
#include <hip/hip_runtime.h>
#include <hip/hip_bf16.h>
#include <stdint.h>

// Problem constants (from the reference)
#define NB 4
#define CH 3
#define HH 512
#define WW 512
#define KK 5
#define PP 2

#define TILE 32
#define TROWS (TILE + KK - 1)       // 36 halo rows
#define TCOLS 40                    // 36 halo cols padded to 40 (10 aligned float4 groups)
#define GRP_PER_ROW (TCOLS / 4)     // 10
#define BLOCK 256                   // 8 wave32s
#define TILES_X (WW / TILE)         // 16
#define TILES_Y (HH / TILE)         // 16
#define NBLOCKS (NB * TILES_X * TILES_Y)  // 1024

// CDNA5 async global->LDS path (gfx1250 only; host pass and other arches fall back)
#if defined(__HIP_DEVICE_COMPILE__) && defined(__gfx1250__)
#define USE_ASYNC_LDS 1
#else
#define USE_ASYNC_LDS 0
#endif

#if USE_ASYNC_LDS
__device__ __forceinline__ void async_copy_b128_to_lds(float* lds_ptr, const float* gptr) {
    // vdst = LDS byte offset (as3 pointer low 32 bits), vaddr = 64-bit global VA, saddr = off (GV mode)
    uint32_t lds_off = (uint32_t)(uintptr_t)(__attribute__((address_space(3))) float*)lds_ptr;
    uint64_t gaddr   = (uint64_t)(uintptr_t)gptr;
    asm volatile("global_load_async_to_lds_b128 %0, %1, off"
                 :: "v"(lds_off), "v"(gaddr)
                 : "memory");
}
#endif

__device__ __forceinline__ int iclamp(int v, int lo, int hi) {
    return v < lo ? lo : (v > hi ? hi : v);
}

__global__ __launch_bounds__(BLOCK)
void cvmin_partial_kernel(const float* __restrict__ pred,
                          const float* __restrict__ target,
                          const float* __restrict__ weight,
                          float* __restrict__ partials) {
    __shared__ float tgt[CH][TROWS][TCOLS];   // 3*36*40*4 = 17280 B

    const int bid = blockIdx.x;
    const int n   = bid / (TILES_X * TILES_Y);
    const int t   = bid % (TILES_X * TILES_Y);
    const int ty0 = (t / TILES_X) * TILE;
    const int tx0 = (t % TILES_X) * TILE;
    const int tid = threadIdx.x;

    const size_t img_base = (size_t)n * CH * HH * WW;

    // ---- Stage target halo tile into LDS as aligned 16B async transfers ----
    // Halo cols span [tx0-4, tx0+36): 10 float4 groups/row, all 16B-aligned.
    // Group starts clamped to [0, WW-4]; the only mis-filled LDS cells map to
    // out-of-image halo positions, which are read solely by border pixels
    // whose weight is forced to zero.
    const int NGRP = CH * TROWS * GRP_PER_ROW;   // 1080
    for (int i = tid; i < NGRP; i += BLOCK) {
        const int c   = i / (TROWS * GRP_PER_ROW);
        const int rem = i % (TROWS * GRP_PER_ROW);
        const int r   = rem / GRP_PER_ROW;
        const int j   = rem % GRP_PER_ROW;
        const int gy  = iclamp(ty0 - PP + r, 0, HH - 1);
        const int gxs = iclamp(tx0 - 4 + 4 * j, 0, WW - 4);
        const float* gp = target + img_base + ((size_t)c * HH + gy) * WW + gxs;
        float* lp = &tgt[c][r][4 * j];
#if USE_ASYNC_LDS
        async_copy_b128_to_lds(lp, gp);
#else
        lp[0] = gp[0]; lp[1] = gp[1]; lp[2] = gp[2]; lp[3] = gp[3];
#endif
    }
#if USE_ASYNC_LDS
    asm volatile("s_wait_asynccnt 0" ::: "memory");
#endif
    __syncthreads();

    // ---- Compute: each thread owns 4 pixels (32x8 layout, 4 row-passes) ----
    const int lx  = tid & (TILE - 1);   // 0..31
    const int ly0 = tid >> 5;           // 0..7
    const size_t cstride = (size_t)HH * WW;

    float loss_acc = 0.0f;
    float w_acc    = 0.0f;

#pragma unroll
    for (int k = 0; k < TILE / 8; ++k) {
        const int ly = ly0 + k * 8;
        const int h  = ty0 + ly;
        const int w  = tx0 + lx;
        const size_t base0 = img_base + (size_t)h * WW + w;  // channel 0

        const float p0 = pred[base0];
        const float p1 = pred[base0 + cstride];
        const float p2 = pred[base0 + 2 * cstride];

        const bool interior = (h >= PP) && (h < HH - PP) && (w >= PP) && (w < WW - PP);
        const float wv = interior ? weight[base0] : 0.0f;   // weight[:, :1]

        // pixel (h,w), offset dx reads gx = w+dx-PP -> LDS col = lx+dx+2 (halo starts at -4)
        float best = 3.0e38f;
#pragma unroll
        for (int dy = 0; dy < KK; ++dy) {
            const float* r0 = &tgt[0][ly + dy][lx + 2];
            const float* r1 = &tgt[1][ly + dy][lx + 2];
            const float* r2 = &tgt[2][ly + dy][lx + 2];
#pragma unroll
            for (int dx = 0; dx < KK; ++dx) {
                float s = fabsf(p0 - r0[dx]) + fabsf(p1 - r1[dx]) + fabsf(p2 - r2[dx]);
                best = fminf(best, s);
            }
        }
        loss_acc += best * (1.0f / 3.0f) * wv;   // mean over C = sum/3
        w_acc    += wv;
    }

    // ---- Deterministic block reduction (wave32 shfl tree + LDS) ----
#pragma unroll
    for (int off = 16; off > 0; off >>= 1) {
        loss_acc += __shfl_down(loss_acc, off, 32);
        w_acc    += __shfl_down(w_acc,    off, 32);
    }
    __shared__ float red[2][BLOCK / 32];
    const int wave = tid >> 5;
    const int lane = tid & 31;
    if (lane == 0) { red[0][wave] = loss_acc; red[1][wave] = w_acc; }
    __syncthreads();
    if (tid == 0) {
        float ls = 0.0f, ws = 0.0f;
#pragma unroll
        for (int i = 0; i < BLOCK / 32; ++i) { ls += red[0][i]; ws += red[1][i]; }
        partials[bid]           = ls;
        partials[NBLOCKS + bid] = ws;
    }
}

__global__ __launch_bounds__(256)
void cvmin_final_kernel(const float* __restrict__ partials, float* __restrict__ out) {
    const int tid = threadIdx.x;
    float ls = 0.0f, ws = 0.0f;
    for (int i = tid; i < NBLOCKS; i += 256) {   // fixed order per thread -> deterministic
        ls += partials[i];
        ws += partials[NBLOCKS + i];
    }
#pragma unroll
    for (int off = 16; off > 0; off >>= 1) {
        ls += __shfl_down(ls, off, 32);
        ws += __shfl_down(ws, off, 32);
    }
    __shared__ float red[2][8];
    const int wave = tid >> 5;
    const int lane = tid & 31;
    if (lane == 0) { red[0][wave] = ls; red[1][wave] = ws; }
    __syncthreads();
    if (tid == 0) {
        float L = 0.0f, Wt = 0.0f;
#pragma unroll
        for (int i = 0; i < 8; ++i) { L += red[0][i]; Wt += red[1][i]; }
        out[0] = L / Wt;   // diff_map.shape[1] == 1, LOSS_WEIGHT == 1.0
    }
}

extern "C" void kernel_launch(void* const* d_in, const int* in_sizes, int n_in,
                              void* d_out, int out_size, void* d_ws, size_t ws_size,
                              hipStream_t stream) {
    (void)in_sizes; (void)n_in; (void)out_size; (void)ws_size;
    const float* pred   = (const float*)d_in[0];
    const float* target = (const float*)d_in[1];
    const float* weight = (const float*)d_in[2];
    float* out      = (float*)d_out;
    float* partials = (float*)d_ws;   // needs 2*NBLOCKS floats = 8 KB

    cvmin_partial_kernel<<<dim3(NBLOCKS), dim3(BLOCK), 0, stream>>>(pred, target, weight, partials);
    cvmin_final_kernel<<<dim3(1), dim3(256), 0, stream>>>(partials, out);
}